// LSTMTagger_68624987455798
// MI455X (gfx1250) — compile-verified
//
#include <hip/hip_runtime.h>
#include <hip/hip_bf16.h>
#include <math.h>

typedef __attribute__((ext_vector_type(2))) float v2f;
typedef __attribute__((ext_vector_type(8))) float v8f;

#define H 100
#define E 100
#define G4 400          // 4*H
#define WN 8192
#define TT 16
#define TOTAL_TOK (WN * TT)   // 131072

// state layout (floats): [0..99]=hF, [100..199]=cF, [200..299]=hB, [300..399]=cB,
//                        [500..599]=hb snapshot (backward output at last word's last char)
#define ST_HB 500
#define ST_HDR 1024

__device__ __forceinline__ float sigmoidf_(float x) {
    return 1.0f / (1.0f + expf(-x));
}

// ---------------------------------------------------------------------------
// Input-projection GEMM for BOTH directions: blockIdx.z = dir.
// Xpre[dir][p, j] = b[j] + sum_e emb(tok_dir(p))[e] * Wih_dir[j][e]
// One wave per 16x16 output tile, V_WMMA_F32_16X16X4_F32 over K=100 (25 steps).
// ---------------------------------------------------------------------------
__global__ __launch_bounds__(32) void inproj_wmma(
    const int* __restrict__ char_ids, const int* __restrict__ lengths,
    const float* __restrict__ char_emb,
    const float* __restrict__ WihF, const float* __restrict__ biasF,
    const float* __restrict__ WihB, const float* __restrict__ biasB,
    float* __restrict__ Xpre, unsigned long long dirStride,
    int chunkStart)
{
    const int mtile = blockIdx.x;          // token tile within chunk
    const int n0    = blockIdx.y * 16;     // gate tile base (0..384)
    const int dir   = blockIdx.z;
    const int lane  = threadIdx.x;
    const int r     = lane & 15;
    const int half  = lane >> 4;

    const float* __restrict__ Wih  = dir ? WihB  : WihF;
    const float* __restrict__ bias = dir ? biasB : biasF;
    float* __restrict__ Xout = Xpre + (size_t)dir * dirStride;

    // Which token does A-row r of this tile correspond to?
    const int tok = chunkStart + mtile * 16 + r;
    int cid;
    if (dir == 0) {
        cid = char_ids[tok];
    } else {
        const int w = tok >> 4, t = tok & 15;
        int rp = lengths[w] - 1 - t;
        if (rp < 0) rp = 0;                 // clipped; masked steps discard result
        cid = char_ids[(w << 4) + rp];
    }
    const float* __restrict__ arow = char_emb + (size_t)cid * E;        // 16B-aligned rows
    const float* __restrict__ brow = Wih + (size_t)(n0 + r) * E;

    v8f acc = {};
    #pragma unroll
    for (int k0 = 0; k0 < E; k0 += 4) {
        const int kk = k0 + 2 * half;       // A/B: VGPR pair holds K={0,1} lo-half, K={2,3} hi-half
        v2f a = *(const v2f*)(arow + kk);   // A[M=r][kk..kk+1]
        v2f b = *(const v2f*)(brow + kk);   // B[kk..kk+1][N=r] = Wih[n0+r][kk..kk+1]
        acc = __builtin_amdgcn_wmma_f32_16x16x4_f32(
            /*neg_a=*/false, a, /*neg_b=*/false, b,
            /*c_mod=*/(short)0, acc, /*reuse_a=*/false, /*reuse_b=*/false);
    }

    const float bj = bias[n0 + r];
    float* outb = Xout + (size_t)(mtile * 16) * G4 + (n0 + r);
    #pragma unroll
    for (int v = 0; v < 8; ++v)             // D: M = v + 8*half, N = r
        outb[(size_t)(v + 8 * half) * G4] = acc[v] + bj;
}

// ---------------------------------------------------------------------------
// Serial LSTM recurrence over one chunk; TWO workgroups run the two
// independent direction chains concurrently (blockIdx.x = dir).
// ---------------------------------------------------------------------------
__global__ __launch_bounds__(512) void lstm_recur(
    const int* __restrict__ lengths,
    const float* __restrict__ WhhF, const float* __restrict__ WhhB,
    const float* __restrict__ Xpre, unsigned long long dirStride,
    float* __restrict__ state,
    int chunkStart, int nTok)
{
    __shared__ __align__(16) float h_s[112];
    __shared__ float c_s[112];
    __shared__ float g_s[G4];

    const int dir = blockIdx.x;
    const int j   = threadIdx.x;
    const float* __restrict__ Whh = dir ? WhhB : WhhF;
    const float* __restrict__ X   = Xpre + (size_t)dir * dirStride;
    float* __restrict__ st = state + dir * 200;   // this direction's carry

    // Each gate-lane keeps its Whh row in 25 float4 registers.
    float4 w4[25];
    if (j < G4) {
        const float4* wp = (const float4*)(Whh + (size_t)j * H);  // rows are 400B -> 16B aligned
        #pragma unroll
        for (int e = 0; e < 25; ++e) w4[e] = wp[e];
    }
    if (j < H) { h_s[j] = st[j]; c_s[j] = st[H + j]; }
    __syncthreads();

    for (int p = 0; p < nTok; ++p) {
        const int tok = chunkStart + p;
        if (j < G4) {
            float acc = X[(size_t)p * G4 + j];
            if (p + 1 < nTok)               // hide next step's read behind update phase
                __builtin_prefetch(&X[(size_t)(p + 1) * G4 + j], 0, 0);
            const float4* hp = (const float4*)h_s;
            #pragma unroll
            for (int e = 0; e < 25; ++e) {
                const float4 hv = hp[e];
                acc += w4[e].x * hv.x + w4[e].y * hv.y + w4[e].z * hv.z + w4[e].w * hv.w;
            }
            g_s[j] = acc;
        }
        __syncthreads();
        if (j < H) {
            const int w = tok >> 4, t = tok & 15;
            if (t < lengths[w]) {           // same mask for fwd and reversed streams
                const float gi = g_s[j], gf = g_s[H + j], gg = g_s[2 * H + j], go = g_s[3 * H + j];
                const float cn = sigmoidf_(gf) * c_s[j] + sigmoidf_(gi) * tanhf(gg);
                const float hn = sigmoidf_(go) * tanhf(cn);
                c_s[j] = cn; h_s[j] = hn;
                // backward output at last word's last char (t==0 of word WN-1)
                if (dir == 1 && tok == ((WN - 1) << 4)) state[ST_HB + j] = hn;
            }
        }
        __syncthreads();
    }
    if (j < H) { st[j] = h_s[j]; st[H + j] = c_s[j]; }
}

__global__ void init_state(float* state, const float* __restrict__ h0,
                           const float* __restrict__ c0)
{
    const int j = threadIdx.x;
    if (j < H) {
        state[j]       = h0[j];       state[H + j]       = c0[j];        // forward carry
        state[200 + j] = h0[H + j];   state[300 + j]     = c0[H + j];    // backward carry
    }
}

// ---------------------------------------------------------------------------
// Sentence BiLSTM (single timestep, both directions) + tag projection.
// hf = forward final h (state[0..99]); hb = snapshot (state[500..599]).
// ---------------------------------------------------------------------------
__global__ __launch_bounds__(512) void sent_and_tag(
    const float* __restrict__ state,
    const float* __restrict__ Wih_sf, const float* __restrict__ Whh_sf, const float* __restrict__ b_sf,
    const float* __restrict__ Wih_sb, const float* __restrict__ Whh_sb, const float* __restrict__ b_sb,
    const float* __restrict__ W_tag, const float* __restrict__ b_tag,
    const float* __restrict__ h0_sent, const float* __restrict__ c0_sent,
    float* __restrict__ out)
{
    __shared__ float feat[2 * H];
    __shared__ float gf_[G4], gb_[G4];
    __shared__ float o_s[2 * H];
    const int j = threadIdx.x;

    if (j < H) { feat[j] = state[j]; feat[H + j] = state[ST_HB + j]; }
    __syncthreads();

    if (j < G4) {
        float af = b_sf[j], ab = b_sb[j];
        const float* rf = Wih_sf + (size_t)j * (2 * H);
        const float* rb = Wih_sb + (size_t)j * (2 * H);
        for (int e = 0; e < 2 * H; ++e) { af += rf[e] * feat[e]; ab += rb[e] * feat[e]; }
        const float* uf = Whh_sf + (size_t)j * H;
        const float* ub = Whh_sb + (size_t)j * H;
        for (int e = 0; e < H; ++e) { af += uf[e] * h0_sent[e]; ab += ub[e] * h0_sent[H + e]; }
        gf_[j] = af; gb_[j] = ab;
    }
    __syncthreads();

    if (j < H) {
        {
            const float gi = gf_[j], gf = gf_[H + j], gg = gf_[2 * H + j], go = gf_[3 * H + j];
            const float cn = sigmoidf_(gf) * c0_sent[j] + sigmoidf_(gi) * tanhf(gg);
            o_s[j] = sigmoidf_(go) * tanhf(cn);
        }
        {
            const float gi = gb_[j], gf = gb_[H + j], gg = gb_[2 * H + j], go = gb_[3 * H + j];
            const float cn = sigmoidf_(gf) * c0_sent[H + j] + sigmoidf_(gi) * tanhf(gg);
            o_s[H + j] = sigmoidf_(go) * tanhf(cn);
        }
    }
    __syncthreads();

    if (j < 32) {
        float a = b_tag[j];
        const float* rw = W_tag + (size_t)j * (2 * H);
        for (int e = 0; e < 2 * H; ++e) a += rw[e] * o_s[e];
        out[j] = a;
    }
}

extern "C" void kernel_launch(void* const* d_in, const int* in_sizes, int n_in,
                              void* d_out, int out_size, void* d_ws, size_t ws_size,
                              hipStream_t stream) {
    const int*   char_ids = (const int*)d_in[0];
    const int*   lengths  = (const int*)d_in[1];
    const float* char_emb = (const float*)d_in[2];
    const float* Wih_cf = (const float*)d_in[3];
    const float* Whh_cf = (const float*)d_in[4];
    const float* b_cf   = (const float*)d_in[5];
    const float* Wih_cb = (const float*)d_in[6];
    const float* Whh_cb = (const float*)d_in[7];
    const float* b_cb   = (const float*)d_in[8];
    const float* Wih_sf = (const float*)d_in[9];
    const float* Whh_sf = (const float*)d_in[10];
    const float* b_sf   = (const float*)d_in[11];
    const float* Wih_sb = (const float*)d_in[12];
    const float* Whh_sb = (const float*)d_in[13];
    const float* b_sb   = (const float*)d_in[14];
    const float* W_tag  = (const float*)d_in[15];
    const float* b_tag  = (const float*)d_in[16];
    const float* h0_char = (const float*)d_in[17];
    const float* c0_char = (const float*)d_in[18];
    const float* h0_sent = (const float*)d_in[19];
    const float* c0_sent = (const float*)d_in[20];
    float* out = (float*)d_out;

    float* wsf   = (float*)d_ws;
    float* state = wsf;
    float* Xpre  = wsf + ST_HDR;   // two per-direction chunk buffers

    // Chunk the hoisted input-projection GEMMs (both dirs) to fit workspace.
    size_t availFloats = (ws_size / sizeof(float) > ST_HDR) ? ws_size / sizeof(float) - ST_HDR : 0;
    long long ct = (long long)(availFloats / (2 * G4));
    ct &= ~15LL;                   // whole 16-token words -> exact WMMA M-tiles
    int chunkTok = (ct > TOTAL_TOK) ? TOTAL_TOK : (int)ct;
    if (chunkTok < 16) chunkTok = 16;
    const unsigned long long dirStride = (unsigned long long)chunkTok * G4;

    hipLaunchKernelGGL(init_state, dim3(1), dim3(128), 0, stream,
                       state, h0_char, c0_char);

    for (int s0 = 0; s0 < TOTAL_TOK; s0 += chunkTok) {
        const int n = (TOTAL_TOK - s0 < chunkTok) ? (TOTAL_TOK - s0) : chunkTok;
        dim3 grid(n / 16, G4 / 16, 2);     // 16-token tiles x 25 gate tiles x 2 dirs
        hipLaunchKernelGGL(inproj_wmma, grid, dim3(32), 0, stream,
                           char_ids, lengths, char_emb,
                           Wih_cf, b_cf, Wih_cb, b_cb,
                           Xpre, dirStride, s0);
        hipLaunchKernelGGL(lstm_recur, dim3(2), dim3(512), 0, stream,
                           lengths, Whh_cf, Whh_cb, Xpre, dirStride, state, s0, n);
    }

    hipLaunchKernelGGL(sent_and_tag, dim3(1), dim3(512), 0, stream,
                       state, Wih_sf, Whh_sf, b_sf, Wih_sb, Whh_sb, b_sb,
                       W_tag, b_tag, h0_sent, c0_sent, out);
}